// DualAdaptiveQuantizer_89781996356236
// MI455X (gfx1250) — compile-verified
//
#include <hip/hip_runtime.h>
#include <hip/hip_bf16.h>

typedef __attribute__((ext_vector_type(2))) float v2f;
typedef __attribute__((ext_vector_type(8))) float v8f;

#define WMMA_F32_16x16x4(A, Bv, C) \
  __builtin_amdgcn_wmma_f32_16x16x4_f32(false, (A), false, (Bv), (short)0, (C), false, false)

static constexpr int B_ = 32, L_ = 1024, K_ = 64, N_ = 8;
static constexpr int M_ = B_ * L_ * K_;                          // 2,097,152 tokens
static constexpr size_t OFF_VQ  = 0;
static constexpr size_t OFF_HQ  = (size_t)M_ * 2;                // 4,194,304
static constexpr size_t OFF_EBD = OFF_HQ + (size_t)M_ * N_ * 2;  // 37,748,736
static constexpr size_t OFF_EBC = OFF_EBD + (size_t)M_;          // 39,845,888
static constexpr size_t OFF_WD  = OFF_EBC + (size_t)M_;          // 41,943,040
static constexpr size_t OFF_WC  = OFF_WD + (size_t)M_ * 3;       // 48,234,496

// ---------------- kernel 1: h_power[b,l,k] = sum_{n,c} H^2 ----------------
__global__ void __launch_bounds__(256) hpower_kernel(const float2* __restrict__ H2,
                                                     float* __restrict__ hp) {
  int t = blockIdx.x * blockDim.x + threadIdx.x;
  if (t >= M_) return;
  int row = t >> 6;          // b*L + l
  int k   = t & 63;
  size_t base = (size_t)row * (N_ * K_) + (size_t)k;
  float s = 0.f;
#pragma unroll
  for (int n = 0; n < N_; ++n) {
    float2 h = H2[base + (size_t)n * K_];
    s += h.x * h.x + h.y * h.y;
  }
  hp[t] = s;
}

// ---------------- kernel 2: avg[b,k] = mean_l h_power (deterministic tree) ----------------
__global__ void __launch_bounds__(256) avg_kernel(const float* __restrict__ hp,
                                                  float* __restrict__ avg) {
  __shared__ float red[256];
  int bk = blockIdx.x;           // b*K + k
  int b = bk >> 6, k = bk & 63;
  float s = 0.f;
  for (int l = (int)threadIdx.x; l < L_; l += 256)
    s += hp[((size_t)(b * L_ + l) << 6) + k];
  red[threadIdx.x] = s;
  __syncthreads();
  for (int off = 128; off > 0; off >>= 1) {
    if ((int)threadIdx.x < off) red[threadIdx.x] += red[threadIdx.x + off];
    __syncthreads();
  }
  if (threadIdx.x == 0) avg[bk] = red[0] * (1.0f / (float)L_);
}

// C-layout -> B-layout chunk builder: one half-swap shuffle + select.
template <int ML, int MH>
__device__ __forceinline__ float mkB(const v8f& c, int lane) {
  float lo, hi;
  if constexpr (ML < 8) lo = c[ML];
  else                  lo = __shfl_xor(c[ML - 8], 16, 32);
  if constexpr (MH >= 8) hi = c[MH - 8];
  else                   hi = __shfl_xor(c[MH], 16, 32);
  return (lane < 16) ? lo : hi;
}

__device__ __forceinline__ float lsqf(float x, float s, float Qn, float Qp) {
  float xs = x / s;
  float xc = fminf(fmaxf(xs, Qn), Qp);
  float xr = xc + (rintf(xc) - xc);   // straight-through round (half-to-even)
  return xr * s;
}

// ---------------- kernel 3: per-token MLP (WMMA) + gumbel + v quant ----------------
__global__ void __launch_bounds__(256) mlp_kernel(
    const float* __restrict__ v, const float* __restrict__ snr,
    const float* __restrict__ gdm, const float* __restrict__ gch,
    const float* __restrict__ W1, const float* __restrict__ b1,
    const float* __restrict__ W2, const float* __restrict__ b2,
    const float* __restrict__ Wd, const float* __restrict__ bd,
    const float* __restrict__ Wc, const float* __restrict__ bc,
    const float* __restrict__ s2d_p, const float* __restrict__ s4d_p,
    const float* __restrict__ hp, const float* __restrict__ avg,
    float* __restrict__ out)
{
  const int lane = (int)(threadIdx.x & 31);
  const int half = lane >> 4;
  const int ml   = lane & 15;

  // ---- preload A fragments (weights transposed: A[m][k] = W[k][m]) ----
  v2f A1[2][2], A2t[2][8], Ah[8];
#pragma unroll
  for (int T = 0; T < 2; ++T) {
    int m = 16 * T + ml;
#pragma unroll
    for (int kc = 0; kc < 2; ++kc)
#pragma unroll
      for (int j = 0; j < 2; ++j) {
        int k = 4 * kc + 2 * half + j;
        A1[T][kc][j] = (k < 5) ? W1[k * 32 + m] : 0.f;
      }
#pragma unroll
    for (int kc = 0; kc < 8; ++kc)
#pragma unroll
      for (int j = 0; j < 2; ++j) {
        int k = 4 * kc + 2 * half + j;
        A2t[T][kc][j] = W2[k * 32 + m];
      }
  }
#pragma unroll
  for (int kc = 0; kc < 8; ++kc)
#pragma unroll
    for (int j = 0; j < 2; ++j) {
      int k = 4 * kc + 2 * half + j;
      float w = 0.f;
      if (ml < 3)      w = Wd[k * 3 + ml];
      else if (ml < 6) w = Wc[k * 3 + (ml - 3)];
      Ah[kc][j] = w;
    }
  float bias1[2][8], bias2[2][8];
#pragma unroll
  for (int T = 0; T < 2; ++T)
#pragma unroll
    for (int r = 0; r < 8; ++r) {
      int m = 16 * T + r + 8 * half;
      bias1[T][r] = b1[m];
      bias2[T][r] = b2[m];
    }
  const float bdv0 = bd[0], bdv1 = bd[1], bdv2 = bd[2];
  const float bcv0 = bc[0], bcv1 = bc[1], bcv2 = bc[2];
  const float s2d = *s2d_p, s4d = *s4d_p;

  const int tiles  = M_ / 16;
  const int nWaves = (int)((gridDim.x * blockDim.x) >> 5);
  const int wid    = (int)((blockIdx.x * blockDim.x + threadIdx.x) >> 5);

  for (int tile = wid; tile < tiles; tile += nWaves) {
    const int t = tile * 16 + ml;   // this lane's token (valid for both halves)
    const float f0 = v[2 * (size_t)t], f1 = v[2 * (size_t)t + 1];
    const float f2 = snr[t], f3 = hp[t];
    const float f4 = avg[((t >> 16) << 6) + (t & 63)];

    // layer-1 B operand: pin^T (5x16, K padded to 8)
    v2f B10, B11;
    B10[0] = half ? f2 : f0;  B10[1] = half ? f3 : f1;
    B11[0] = half ? 0.f : f4; B11[1] = 0.f;

    v8f c0 = {}, c1 = {};
    c0 = WMMA_F32_16x16x4(A1[0][0], B10, c0);
    c0 = WMMA_F32_16x16x4(A1[0][1], B11, c0);
    c1 = WMMA_F32_16x16x4(A1[1][0], B10, c1);
    c1 = WMMA_F32_16x16x4(A1[1][1], B11, c1);
#pragma unroll
    for (int r = 0; r < 8; ++r) {
      c0[r] = fmaxf(c0[r] + bias1[0][r], 0.f);
      c1[r] = fmaxf(c1[r] + bias1[1][r], 0.f);
    }

    // layer 2: K=32 (8 chunks)
    v8f d0 = {}, d1 = {};
#define L2STEP(CT, MB, KC) { v2f Bv;                           \
      Bv[0] = mkB<(MB), (MB) + 2>(CT, lane);                   \
      Bv[1] = mkB<(MB) + 1, (MB) + 3>(CT, lane);               \
      d0 = WMMA_F32_16x16x4(A2t[0][KC], Bv, d0);               \
      d1 = WMMA_F32_16x16x4(A2t[1][KC], Bv, d1); }
    L2STEP(c0, 0, 0)  L2STEP(c0, 4, 1)  L2STEP(c0, 8, 2)  L2STEP(c0, 12, 3)
    L2STEP(c1, 0, 4)  L2STEP(c1, 4, 5)  L2STEP(c1, 8, 6)  L2STEP(c1, 12, 7)
#undef L2STEP
#pragma unroll
    for (int r = 0; r < 8; ++r) {
      d0[r] = fmaxf(d0[r] + bias2[0][r], 0.f);
      d1[r] = fmaxf(d1[r] + bias2[1][r], 0.f);
    }

    // heads: [Wd|Wc]^T (6 rows used, padded to 16)
    v8f hE = {};
#define HSTEP(CT, MB, KC) { v2f Bv;                            \
      Bv[0] = mkB<(MB), (MB) + 2>(CT, lane);                   \
      Bv[1] = mkB<(MB) + 1, (MB) + 3>(CT, lane);               \
      hE = WMMA_F32_16x16x4(Ah[KC], Bv, hE); }
    HSTEP(d0, 0, 0)  HSTEP(d0, 4, 1)  HSTEP(d0, 8, 2)  HSTEP(d0, 12, 3)
    HSTEP(d1, 0, 4)  HSTEP(d1, 4, 5)  HSTEP(d1, 8, 6)  HSTEP(d1, 12, 7)
#undef HSTEP

    // lanes 0..15 hold the 6 logits of their token in hE[0..5]
    const float gd0 = gdm[3 * (size_t)t], gd1 = gdm[3 * (size_t)t + 1], gd2 = gdm[3 * (size_t)t + 2];
    const float gc0 = gch[3 * (size_t)t], gc1 = gch[3 * (size_t)t + 1], gc2 = gch[3 * (size_t)t + 2];

    // demod gumbel-softmax (tau = 1)
    float x0 = (hE[0] + bdv0) + gd0, x1 = (hE[1] + bdv1) + gd1, x2 = (hE[2] + bdv2) + gd2;
    float mx = fmaxf(x0, fmaxf(x1, x2));
    float p0 = expf(x0 - mx), p1 = expf(x1 - mx), p2 = expf(x2 - mx);
    float ps = p0 + p1 + p2;
    float y0 = p0 / ps, y1 = p1 / ps, y2 = p2 / ps;
    int ad = 0; float bv = y0;
    if (y1 > bv) { ad = 1; bv = y1; }
    if (y2 > bv) { ad = 2; }
    float wd0 = ((ad == 0 ? 1.f : 0.f) + y0) - y0;
    float wd1 = ((ad == 1 ? 1.f : 0.f) + y1) - y1;
    float wd2 = ((ad == 2 ? 1.f : 0.f) + y2) - y2;

    // channel gumbel-softmax
    x0 = (hE[3] + bcv0) + gc0; x1 = (hE[4] + bcv1) + gc1; x2 = (hE[5] + bcv2) + gc2;
    mx = fmaxf(x0, fmaxf(x1, x2));
    p0 = expf(x0 - mx); p1 = expf(x1 - mx); p2 = expf(x2 - mx);
    ps = p0 + p1 + p2;
    y0 = p0 / ps; y1 = p1 / ps; y2 = p2 / ps;
    int ac = 0; bv = y0;
    if (y1 > bv) { ac = 1; bv = y1; }
    if (y2 > bv) { ac = 2; }
    float wc0 = ((ac == 0 ? 1.f : 0.f) + y0) - y0;
    float wc1 = ((ac == 1 ? 1.f : 0.f) + y1) - y1;
    float wc2 = ((ac == 2 ? 1.f : 0.f) + y2) - y2;

    if (lane < 16) {
      const size_t st = (size_t)t;
      out[OFF_WD + 3 * st + 0] = wd0;
      out[OFF_WD + 3 * st + 1] = wd1;
      out[OFF_WD + 3 * st + 2] = wd2;
      out[OFF_WC + 3 * st + 0] = wc0;
      out[OFF_WC + 3 * st + 1] = wc1;
      out[OFF_WC + 3 * st + 2] = wc2;
      out[OFF_EBD + st] = wd1 * 4.f + wd2 * 8.f;
      out[OFF_EBC + st] = wc1 * 32.f + wc2 * 64.f;   // 2*N*2=32, 2*N*4=64 (N=8)
      out[OFF_VQ + 2 * st + 0] =
          wd1 * lsqf(f0, s2d, -2.f, 1.f) + wd2 * lsqf(f0, s4d, -8.f, 7.f);
      out[OFF_VQ + 2 * st + 1] =
          wd1 * lsqf(f1, s2d, -2.f, 1.f) + wd2 * lsqf(f1, s4d, -8.f, 7.f);
    }
  }
}

// ---------------- kernel 4: H_q (reads w_channel from d_out) ----------------
__global__ void __launch_bounds__(256) hq_kernel(const float2* __restrict__ H2,
                                                 const float* __restrict__ s2c_p,
                                                 const float* __restrict__ s4c_p,
                                                 float* __restrict__ out) {
  int t = blockIdx.x * blockDim.x + threadIdx.x;
  if (t >= M_) return;
  const float s2c = *s2c_p, s4c = *s4c_p;
  const float wc1 = out[OFF_WC + 3 * (size_t)t + 1];
  const float wc2 = out[OFF_WC + 3 * (size_t)t + 2];
  float2* Hq2 = (float2*)(out + OFF_HQ);
  size_t base = (size_t)(t >> 6) * (N_ * K_) + (size_t)(t & 63);
#pragma unroll
  for (int n = 0; n < N_; ++n) {
    float2 h = H2[base + (size_t)n * K_];
    float2 q;
    q.x = wc1 * lsqf(h.x, s2c, -2.f, 1.f) + wc2 * lsqf(h.x, s4c, -8.f, 7.f);
    q.y = wc1 * lsqf(h.y, s2c, -2.f, 1.f) + wc2 * lsqf(h.y, s4c, -8.f, 7.f);
    Hq2[base + (size_t)n * K_] = q;
  }
}

extern "C" void kernel_launch(void* const* d_in, const int* in_sizes, int n_in,
                              void* d_out, int out_size, void* d_ws, size_t ws_size,
                              hipStream_t stream) {
  (void)in_sizes; (void)n_in; (void)out_size; (void)ws_size;
  const float* v   = (const float*)d_in[0];
  const float* H   = (const float*)d_in[1];
  const float* snr = (const float*)d_in[2];
  const float* gd  = (const float*)d_in[3];
  const float* gc  = (const float*)d_in[4];
  const float* W1  = (const float*)d_in[5];
  const float* b1  = (const float*)d_in[6];
  const float* W2  = (const float*)d_in[7];
  const float* b2  = (const float*)d_in[8];
  const float* Wd  = (const float*)d_in[9];
  const float* bd  = (const float*)d_in[10];
  const float* Wc  = (const float*)d_in[11];
  const float* bc  = (const float*)d_in[12];
  const float* s2d = (const float*)d_in[13];
  const float* s4d = (const float*)d_in[14];
  const float* s2c = (const float*)d_in[15];
  const float* s4c = (const float*)d_in[16];
  float* out = (float*)d_out;
  float* avg = (float*)d_ws;           // 2048 floats (8 KB)
  float* hp  = out + OFF_HQ;           // stash h_power in not-yet-written H_q region

  hpower_kernel<<<M_ / 256, 256, 0, stream>>>((const float2*)H, hp);
  avg_kernel<<<B_ * K_, 256, 0, stream>>>(hp, avg);
  mlp_kernel<<<1024, 256, 0, stream>>>(v, snr, gd, gc, W1, b1, W2, b2,
                                       Wd, bd, Wc, bc, s2d, s4d, hp, avg, out);
  hq_kernel<<<M_ / 256, 256, 0, stream>>>((const float2*)H, s2c, s4c, out);
}